// Conv2D_35210141892837
// MI455X (gfx1250) — compile-verified
//
#include <hip/hip_runtime.h>

// ---------------------------------------------------------------------------
// 3x3 VALID conv (8192x8192 f32 -> 8190x8190 f32) + bias, via CDNA5 WMMA.
// Wave computes 8 x (16x16) output tiles (16 rows x 128 cols) with
// v_wmma_f32_16x16x4_f32 using a banded-Toeplitz B built ONCE per wave and
// pinned in VGPRs (asm) so the compiler cannot rematerialize the selects.
// WG = 8 waves = 16 x 1024 outputs; input staged once per WG through LDS
// with async global->LDS b128 loads (ASYNCcnt + barrier).
// ---------------------------------------------------------------------------

typedef float v2f __attribute__((ext_vector_type(2)));
typedef float v8f __attribute__((ext_vector_type(8)));
typedef int   v4i __attribute__((ext_vector_type(4)));

#define GLOBAL_AS __attribute__((address_space(1)))
#define LDS_AS    __attribute__((address_space(3)))

#define HAS_ASYNC_LDS (__has_builtin(__builtin_amdgcn_global_load_async_to_lds_b128) && \
                       __has_builtin(__builtin_amdgcn_global_load_async_to_lds_b32)  && \
                       __has_builtin(__builtin_amdgcn_s_wait_asynccnt))

namespace {
constexpr int W     = 8192;   // input width/height
constexpr int OW    = 8190;   // output width/height
constexpr int PITCH = 1028;   // LDS row pitch in floats (1028 % 64 == 4 -> conflict-free)
constexpr int ROWS  = 18;     // 16 output rows + 2 halo rows
constexpr int TPW   = 8;      // 16x16 tiles per wave (wave covers 16 x 128 outputs)
}

__device__ __forceinline__ int imin(int a, int b) { return a < b ? a : b; }

// select w[j] for j in {0,1,2}, else 0 — compiles to v_cndmask chain
__device__ __forceinline__ float selw(int j, float w0, float w1, float w2) {
    float r = 0.0f;
    r = (j == 0) ? w0 : r;
    r = (j == 1) ? w1 : r;
    r = (j == 2) ? w2 : r;
    return r;
}

__global__ __launch_bounds__(256)
void conv3x3_wmma_kernel(const float* __restrict__ xg,
                         const float* __restrict__ wg,
                         const float* __restrict__ bg,
                         float* __restrict__ outg)
{
    __shared__ float smem[ROWS * PITCH];

    const int tid   = threadIdx.x;
    const int xbase = blockIdx.x * 1024;  // first input/output column of this WG
    const int y0    = blockIdx.y * 16;    // first output row of this WG

    // ---- stage 18 x 1028 input block into LDS (async b128 path) ----
    constexpr int NV4 = ROWS * (PITCH / 4);     // 4626 float4 slots
    for (int i4 = tid; i4 < NV4; i4 += 256) {
        const int r   = i4 / (PITCH / 4);
        const int c4  = (i4 % (PITCH / 4)) * 4;
        const int gr  = imin(y0 + r, W - 1);    // clamp halo rows at bottom edge
        const int gc  = xbase + c4;
        float* lp = &smem[r * PITCH + c4];
        if (gc + 3 <= W - 1) {
            const size_t go = (size_t)gr * W + gc;
#if HAS_ASYNC_LDS
            __builtin_amdgcn_global_load_async_to_lds_b128(
                (GLOBAL_AS v4i*)(xg + go),
                (LDS_AS v4i*)lp, 0, 0);
#else
            const float4 v = *(const float4*)(xg + go);
            *(float4*)lp = v;
#endif
        } else {
            // right-edge block: clamp per element (values are killed by zero taps)
#pragma unroll
            for (int e = 0; e < 4; ++e) {
                const int    gce = imin(gc + e, W - 1);
                const size_t go  = (size_t)gr * W + gce;
#if HAS_ASYNC_LDS
                __builtin_amdgcn_global_load_async_to_lds_b32(
                    (GLOBAL_AS int*)(xg + go),
                    (LDS_AS int*)(lp + e), 0, 0);
#else
                lp[e] = xg[go];
#endif
            }
        }
    }

    // ---- per-thread constants / B fragments (overlap with async loads) ----
    const float w00 = wg[0], w01 = wg[1], w02 = wg[2];
    const float w10 = wg[3], w11 = wg[4], w12 = wg[5];
    const float w20 = wg[6], w21 = wg[7], w22 = wg[8];
    const float bias = bg[0];

    const int lane = tid & 31;
    const int wave = tid >> 5;
    const int half = lane >> 4;   // 0: lanes 0-15, 1: lanes 16-31
    const int mn   = lane & 15;   // M for A / N for B,C,D
    const int kb   = half * 2;    // K base for this half-wave
    const int xw   = wave * 128;  // this wave's column offset inside the LDS block

    // B[k,n] = w[i, 4t+k-n] (banded Toeplitz); identical for every tile & wave.
    // Built once (15 v2f fragments) and PINNED in VGPRs via empty asm so the
    // compiler cannot rematerialize the cndmask chains inside the tile loop.
    v2f B[3][5];
#pragma unroll
    for (int t = 0; t < 5; ++t) {
        const int j0 = 4 * t + kb - mn;
#pragma unroll
        for (int i = 0; i < 3; ++i) {
            const float wr0 = (i == 0) ? w00 : (i == 1) ? w10 : w20;
            const float wr1 = (i == 0) ? w01 : (i == 1) ? w11 : w21;
            const float wr2 = (i == 0) ? w02 : (i == 1) ? w12 : w22;
            float bx = selw(j0,     wr0, wr1, wr2);
            float by = selw(j0 + 1, wr0, wr1, wr2);
            asm("" : "+v"(bx), "+v"(by));   // opaque: forces B into live VGPRs
            B[i][t].x = bx;
            B[i][t].y = by;
        }
    }

#if HAS_ASYNC_LDS
    __builtin_amdgcn_s_wait_asynccnt(0);
#endif
    __syncthreads();

    // per-lane A base: row mn, wave column origin, K-base; all else is
    // compile-time immediate offsets -> ds_loads use immediate addressing.
    const float* rbase = &smem[mn * PITCH + xw + kb];

    const int  orow0    = y0 + half * 8;
    const bool rows_ok  = (y0 + 16 <= OW);          // uniform: no bottom clipping

    // ---- 8 tiles x 15 WMMAs ----
#pragma unroll
    for (int tt = 0; tt < TPW; ++tt) {
        v8f acc;
#pragma unroll
        for (int r = 0; r < 8; ++r) acc[r] = bias;

#pragma unroll
        for (int i = 0; i < 3; ++i) {
#pragma unroll
            for (int t = 0; t < 5; ++t) {
                v2f a;                 // A[m,k] = x[y0+m+i, xbase+xw+tt*16+4t+k]
                a.x = rbase[i * PITCH + tt * 16 + 4 * t];
                a.y = rbase[i * PITCH + tt * 16 + 4 * t + 1];
                acc = __builtin_amdgcn_wmma_f32_16x16x4_f32(
                    false, a, false, B[i][t], (short)0, acc, false, false);
            }
        }

        // store 16x16 tile (C/D layout: vgpr r -> M = r + 8*half, N = mn)
        const int    ocol0 = xbase + xw + tt * 16;       // uniform per tile
        const int    ocol  = ocol0 + mn;
        const size_t obase = (size_t)orow0 * OW + ocol;
        if (rows_ok && (ocol0 + 15 < OW)) {
            // fast path: whole tile in bounds, no per-row compares
#pragma unroll
            for (int r = 0; r < 8; ++r)
                outg[obase + (size_t)r * OW] = acc[r];
        } else if (ocol < OW) {
            // edge tiles (bx==7 last wave / by==511): guarded stores
#pragma unroll
            for (int r = 0; r < 8; ++r) {
                if (orow0 + r < OW)
                    outg[obase + (size_t)r * OW] = acc[r];
            }
        }
    }
}

extern "C" void kernel_launch(void* const* d_in, const int* in_sizes, int n_in,
                              void* d_out, int out_size, void* d_ws, size_t ws_size,
                              hipStream_t stream)
{
    (void)in_sizes; (void)n_in; (void)out_size; (void)d_ws; (void)ws_size;
    const float* x    = (const float*)d_in[0];
    const float* wgt  = (const float*)d_in[1];
    const float* bias = (const float*)d_in[2];
    float* out        = (float*)d_out;

    // WG (8 waves) covers 16 rows x 1024 cols of output; 8 x 512 WGs total.
    dim3 grid(8, 512, 1);
    dim3 block(256, 1, 1);
    conv3x3_wmma_kernel<<<grid, block, 0, stream>>>(x, wgt, bias, out);
}